// ParabolicPool2D_V2_40106404610419
// MI455X (gfx1250) — compile-verified
//
#include <hip/hip_runtime.h>

// Morphological dilation (max-plus pool), KS=7, STRIDE=2, per-channel parabolic kernel.
// f: [B,96,224,224] f32, t: [96] f32 -> out: [B,96,112,112] f32.
// Memory-bound (385 MB @ 23.3 TB/s ~= 16.5 us floor). CDNA5 async global->LDS staging
// (ASYNCcnt) + separable max-plus factorization (49 taps -> 7+7 taps).

#define KS        7
#define CCH       96
#define HI        224
#define WI        224
#define HO        112
#define WO        112
#define TILE_OY   16
#define IN_ROWS   (2 * TILE_OY + 5)   // 37 input rows per tile
#define IN_STRIDE 232                  // padded LDS row (floats): 1 pad chunk + 56 data + 1 pad
#define NTHREADS  256

#if defined(__AMDGCN__) && __has_builtin(__builtin_amdgcn_global_load_async_to_lds_b128)
#define USE_ASYNC_LDS 1
#else
#define USE_ASYNC_LDS 0
#endif

// Builtin prototype (probe-confirmed): (v4i AS1*, v4i AS3*, imm int offset, imm int cpol)
typedef int v4i_t __attribute__((ext_vector_type(4)));
typedef __attribute__((address_space(1))) v4i_t* g_v4i_ptr;
typedef __attribute__((address_space(3))) v4i_t* l_v4i_ptr;

__global__ __launch_bounds__(NTHREADS)
void parabolic_pool2d_kernel(const float* __restrict__ f,
                             const float* __restrict__ t,
                             float* __restrict__ out)
{
    __shared__ float s_in[IN_ROWS * IN_STRIDE];  // input tile, -inf padded borders
    __shared__ float s_g[IN_ROWS * WO];          // horizontal-pass intermediate
    __shared__ float s_a[KS];                    // separable additive taps

    const int ty  = blockIdx.x;   // 0..6  (output-row tile)
    const int c   = blockIdx.y;   // channel
    const int b   = blockIdx.z;   // batch
    const int tid = threadIdx.x;

    // a[k] = -z[k]^2 / (4 t[c]),  z = linspace(-2, 3, 7) = -2 + (5/6) k
    if (tid < KS) {
        const float z = -2.0f + (5.0f / 6.0f) * (float)tid;
        s_a[tid] = -(z * z) / (4.0f * t[c]);
    }

    // ---- stage tile: pad/OOB slots get -inf, valid data slots get async copies.
    // The two sets are disjoint, so one barrier (after s_wait_asynccnt) suffices.
    const float NEG_INF = -__builtin_inff();
    const float4 ninf4 = make_float4(NEG_INF, NEG_INF, NEG_INF, NEG_INF);
    const float* fplane = f + (size_t)(b * CCH + c) * (HI * WI);
    const int y0in = ty * (2 * TILE_OY) - (KS / 2);

    const int ROW_CHUNKS = IN_STRIDE / 4;  // 58: [0]=left pad, [1..56]=data, [57]=right pad
    for (int task = tid; task < IN_ROWS * ROW_CHUNKS; task += NTHREADS) {
        const int r  = task / ROW_CHUNKS;
        const int ch = task - r * ROW_CHUNKS;
        const int y  = y0in + r;
        float* dst = &s_in[r * IN_STRIDE + ch * 4];
        if (((unsigned)y < (unsigned)HI) && (ch >= 1) && (ch <= WI / 4)) {
            const float* src = fplane + (size_t)y * WI + (ch - 1) * 4;
#if USE_ASYNC_LDS
            __builtin_amdgcn_global_load_async_to_lds_b128(
                (g_v4i_ptr)src, (l_v4i_ptr)dst, /*offset=*/0, /*cpol=*/0);
#else
            *(float4*)dst = *(const float4*)src;
#endif
        } else {
            *(float4*)dst = ninf4;  // -inf padding (borders and OOB rows)
        }
    }
#if USE_ASYNC_LDS && __has_builtin(__builtin_amdgcn_s_wait_asynccnt)
    __builtin_amdgcn_s_wait_asynccnt(0);
#endif
    __syncthreads();

    // ---- horizontal max-plus pass: g[r][ox] = max_k in[r][2ox+k-3] + a[k] ----
    for (int i = tid; i < IN_ROWS * WO; i += NTHREADS) {
        const int r  = i / WO;
        const int ox = i - r * WO;
        const float* row = &s_in[r * IN_STRIDE + 2 * ox + 1];  // 4 + (2ox - 3)
        float m = NEG_INF;
#pragma unroll
        for (int k = 0; k < KS; ++k)
            m = fmaxf(m, row[k] + s_a[k]);
        s_g[i] = m;
    }
    __syncthreads();

    // ---- vertical max-plus pass + coalesced store ----
    float* oplane = out + (size_t)(b * CCH + c) * (HO * WO) + (size_t)(ty * TILE_OY) * WO;
    for (int i = tid; i < TILE_OY * WO; i += NTHREADS) {
        const int oy = i / WO;
        const int ox = i - oy * WO;
        const float* gcol = &s_g[(2 * oy) * WO + ox];
        float m = NEG_INF;
#pragma unroll
        for (int k = 0; k < KS; ++k)
            m = fmaxf(m, gcol[k * WO] + s_a[k]);
        oplane[i] = m;
    }
}

extern "C" void kernel_launch(void* const* d_in, const int* in_sizes, int n_in,
                              void* d_out, int out_size, void* d_ws, size_t ws_size,
                              hipStream_t stream) {
    const float* f = (const float*)d_in[0];
    const float* t = (const float*)d_in[1];
    float* out = (float*)d_out;

    const int B = in_sizes[0] / (CCH * HI * WI);  // 16 for the reference shapes
    dim3 grid(HO / TILE_OY, CCH, B);              // (7, 96, B)
    parabolic_pool2d_kernel<<<grid, NTHREADS, 0, stream>>>(f, t, out);
}